// KNNBuffer_74062416053213
// MI455X (gfx1250) — compile-verified
//
#include <hip/hip_runtime.h>
#include <stdint.h>

#define B_Q    2048
#define N_ROWS 100000
#define D_DIM  512
#define MTILE  64      // queries per block = 4 waves x 16
#define GRID_Y 125     // 6250 n-tiles total, 50 per block
#define NT_PER_BLK 50

typedef float v2f __attribute__((ext_vector_type(2)));
typedef float v8f __attribute__((ext_vector_type(8)));

// Order-preserving float->uint encoding (unsigned compare == float compare).
__device__ __forceinline__ unsigned int enc_f32(float f) {
    unsigned int u = __float_as_uint(f);
    return (u & 0x80000000u) ? ~u : (u | 0x80000000u);
}

__device__ __forceinline__ unsigned long long shfl_xor_u64(unsigned long long v, int mask) {
    unsigned int lo = (unsigned int)v;
    unsigned int hi = (unsigned int)(v >> 32);
    lo = __shfl_xor(lo, mask, 32);
    hi = __shfl_xor(hi, mask, 32);
    return ((unsigned long long)hi << 32) | lo;
}

// Async copy of one 16x512 f32 tile (32KB) from global to LDS.
// 128 threads x 16B x 16 slices; INST_OFFSET applies to both LDS and
// global addresses, so one lds-addr VGPR + one global-offset VGPR suffice.
__device__ __forceinline__ void fill_tile_async(uint32_t lds_base,
                                                const char* gbase, int tid) {
    const uint32_t lds  = lds_base + (uint32_t)tid * 16u;
    const uint32_t goff = (uint32_t)tid * 16u;
#pragma unroll
    for (int j = 0; j < 16; ++j) {
        asm volatile("global_load_async_to_lds_b128 %0, %1, %2 offset:%3"
                     :: "v"(lds), "v"(goff), "s"(gbase), "n"(j * 2048)
                     : "memory");
    }
}

__global__ void knn_init(unsigned long long* best) {
    int i = blockIdx.x * blockDim.x + threadIdx.x;
    if (i < B_Q) best[i] = 0xFFFFFFFFFFFFFFFFull;
}

// ||buffer_n||^2, one wave (32 lanes) per row.
__global__ void knn_b2(const float* __restrict__ buf, float* __restrict__ b2) {
    const int row  = blockIdx.x * 8 + (threadIdx.x >> 5);
    const int lane = threadIdx.x & 31;
    if (row >= N_ROWS) return;
    const float4* p = (const float4*)(buf + (size_t)row * D_DIM);
    float s = 0.f;
#pragma unroll
    for (int j = 0; j < 4; ++j) {
        float4 v = p[lane + 32 * j];
        s = fmaf(v.x, v.x, s); s = fmaf(v.y, v.y, s);
        s = fmaf(v.z, v.z, s); s = fmaf(v.w, v.w, s);
    }
#pragma unroll
    for (int m = 1; m < 32; m <<= 1) s += __shfl_xor(s, m, 32);
    if (lane == 0) b2[row] = s;
}

// Main: fp32 WMMA GEMM tiles, async double-buffered LDS staging,
// packed (dist,idx) running argmin with u64 atomic min.
__global__ __launch_bounds__(128) void knn_main(const float* __restrict__ q,
                                                const float* __restrict__ buf,
                                                const float* __restrict__ b2,
                                                unsigned long long* __restrict__ best) {
    extern __shared__ float smem[];
    float* btile = smem;                        // [2][16][512]  2 x 32KB
    float* qtile = smem + 2 * 16 * D_DIM;       // [64][512]     128KB

    const int tid  = threadIdx.x;
    const int lane = tid & 31;
    const int wave = tid >> 5;
    const int half = lane >> 4;   // selects K pair {0,1} vs {2,3} inside 16x4 step
    const int nl   = lane & 15;

    const int qbase = blockIdx.x * MTILE;
    const int t0    = blockIdx.y * NT_PER_BLK;

    // LDS byte offsets of the two btile buffers (low 32 bits of flat addr).
    const uint32_t bt_lds0 = (uint32_t)(uintptr_t)btile;
    const uint32_t bt_lds1 = bt_lds0 + 16 * D_DIM * sizeof(float);

    // Kick off the async fill of tile t0 into buffer 0 immediately.
    fill_tile_async(bt_lds0, (const char*)buf + (size_t)t0 * 32768, tid);

    // Stage the 64x512 query tile (contiguous in memory) into LDS, coalesced.
    {
        const float4* src = (const float4*)(q + (size_t)qbase * D_DIM);
        float4* dst = (float4*)qtile;
        for (int i = tid; i < MTILE * D_DIM / 4; i += 128) dst[i] = src[i];
    }

    const float* qrow = qtile + (wave * 16 + nl) * D_DIM + half * 2;

    unsigned long long bestv[8];
#pragma unroll
    for (int r = 0; r < 8; ++r) bestv[r] = 0xFFFFFFFFFFFFFFFFull;

    for (int t = 0; t < NT_PER_BLK; ++t) {
        const int cur   = t & 1;
        const int nrow0 = (t0 + t) * 16;

        // All waves done reading btile[1-cur] (compute of t-1) before refill.
        __syncthreads();
        if (t + 1 < NT_PER_BLK) {
            fill_tile_async(cur ? bt_lds0 : bt_lds1,
                            (const char*)buf + (size_t)(t0 + t + 1) * 32768, tid);
            // Allow the 16 in-flight ops of tile t+1; tile t's fill is complete.
            asm volatile("s_wait_asynccnt 16" ::: "memory");
        } else {
            asm volatile("s_wait_asynccnt 0" ::: "memory");
        }
        __syncthreads();   // every wave's fill of tile t is now visible

        const int nidx = nrow0 + nl;            // lane's N column
        const float bb = b2[nidx];              // hoisted: hides under 128 WMMAs

        const float* brow = btile + cur * (16 * D_DIM) + nl * D_DIM + half * 2;
        v8f acc = {};
#pragma unroll 8
        for (int kk = 0; kk < D_DIM; kk += 4) {
            v2f a = *(const v2f*)(qrow + kk);   // A[m][kk+off], A[m][kk+off+1]
            v2f b = *(const v2f*)(brow + kk);   // B[kk+off][n], B[kk+off+1][n]
            acc = __builtin_amdgcn_wmma_f32_16x16x4_f32(
                false, a, false, b, (short)0, acc, false, false);
        }

#pragma unroll
        for (int r = 0; r < 8; ++r) {
            float key = fmaf(-2.0f, acc[r], bb);          // ||b||^2 - 2 q.b
            unsigned long long cand =
                ((unsigned long long)enc_f32(key) << 32) | (unsigned int)nidx;
            bestv[r] = bestv[r] < cand ? bestv[r] : cand;
        }
    }

    // Min-reduce over the 16 lanes sharing each M-group (xor 1..8 stays in-group).
#pragma unroll
    for (int r = 0; r < 8; ++r) {
#pragma unroll
        for (int m = 1; m <= 8; m <<= 1) {
            unsigned long long o = shfl_xor_u64(bestv[r], m);
            bestv[r] = bestv[r] < o ? bestv[r] : o;
        }
    }
    if (nl == 0) {
        const int mbase = qbase + wave * 16 + half * 8;   // C layout: M = r + 8*half
#pragma unroll
        for (int r = 0; r < 8; ++r)
            atomicMin(&best[mbase + r], bestv[r]);
    }
}

// Gather winning buffer rows to output; one wave per query.
__global__ void knn_gather(const float* __restrict__ buf,
                           const unsigned long long* __restrict__ best,
                           float* __restrict__ out) {
    const int qi   = blockIdx.x * 8 + (threadIdx.x >> 5);
    const int lane = threadIdx.x & 31;
    if (qi >= B_Q) return;
    const unsigned int idx = (unsigned int)(best[qi] & 0xFFFFFFFFu);
    const float4* src = (const float4*)(buf + (size_t)idx * D_DIM);
    float4* dst = (float4*)(out + (size_t)qi * D_DIM);
#pragma unroll
    for (int j = 0; j < 4; ++j) dst[lane + 32 * j] = src[lane + 32 * j];
}

extern "C" void kernel_launch(void* const* d_in, const int* in_sizes, int n_in,
                              void* d_out, int out_size, void* d_ws, size_t ws_size,
                              hipStream_t stream) {
    const float* q   = (const float*)d_in[0];   // [2048, 512]
    const float* buf = (const float*)d_in[1];   // [100000, 512]
    float* out = (float*)d_out;                 // [2048, 512]

    unsigned long long* best = (unsigned long long*)d_ws;                    // 16 KB
    float* b2 = (float*)((char*)d_ws + B_Q * sizeof(unsigned long long));    // 400 KB

    knn_init<<<(B_Q + 255) / 256, 256, 0, stream>>>(best);
    knn_b2<<<(N_ROWS + 7) / 8, 256, 0, stream>>>(buf, b2);

    const size_t shmem = (size_t)(2 * 16 * D_DIM + MTILE * D_DIM) * sizeof(float); // 192 KB
    (void)hipFuncSetAttribute((const void*)knn_main,
                              hipFuncAttributeMaxDynamicSharedMemorySize, (int)shmem);
    dim3 grid(B_Q / MTILE, GRID_Y);
    knn_main<<<grid, 128, shmem, stream>>>(q, buf, b2, best);

    knn_gather<<<(B_Q + 7) / 8, 256, 0, stream>>>(buf, best, out);
}